// Policy_loss_54752243089924
// MI455X (gfx1250) — compile-verified
//
#include <hip/hip_runtime.h>

typedef __attribute__((ext_vector_type(2))) float v2f;
typedef __attribute__((ext_vector_type(8))) float v8f;

#define BETA 0.1f

// Kernel 1: one wave32 per row of `action`.
// Each float4 load is exactly one size-4 segment -> fmax3 -> accumulate.
// Lanes stride by 32 segments => consecutive lanes read consecutive 16B
// (512B per wave load instruction, perfectly coalesced for HBM streaming).
__global__ void rowReduceKernel(const float* __restrict__ action,
                                const float* __restrict__ reward,
                                float* __restrict__ partials,
                                int m, int B)
{
    const int wave = threadIdx.x >> 5;      // 8 waves per block
    const int lane = threadIdx.x & 31;
    const int row  = blockIdx.x * 8 + wave;

    float s = 0.0f;
    if (row < B) {
        const float4* p = (const float4*)(action + (size_t)row * (size_t)(4 * m));
        #pragma unroll 4
        for (int i = lane; i < m; i += 32) {
            // gfx1250 global_prefetch_b8 for the next chunk (speculative; OOB dropped)
            __builtin_prefetch(p + i + 64, 0, 0);
            float4 v = p[i];
            s += fmaxf(fmaxf(v.x, v.y), fmaxf(v.z, v.w));
        }
    }

    // wave32 cross-lane sum (fixed order -> deterministic)
    #pragma unroll
    for (int off = 16; off > 0; off >>= 1) s += __shfl_xor(s, off, 32);

    __shared__ float ls[8];
    if (lane == 0) {
        float c = 0.0f;
        if (row < B) {
            float a_n = s / (float)m;          // mean of segment maxes
            float l   = logf(a_n) * a_n;       // log(a_n) * a_n
            c = l * (reward[row] + BETA);      // l*r + l*beta contribution
        }
        ls[wave] = c;
    }
    __syncthreads();
    if (threadIdx.x == 0) {
        float acc = 0.0f;
        #pragma unroll
        for (int i = 0; i < 8; ++i) acc += ls[i];
        partials[blockIdx.x] = acc;            // one partial per block
    }
}

// Kernel 2: single wave32 (EXEC guaranteed all-ones, as WMMA requires).
// Serial strided accumulation of block partials into 32 lane values, then a
// v_wmma_f32_16x16x4_f32 with B = ones performs the 32-lane sum exactly in f32:
//   A[i][0] = v_i (lanes 0-15, VGPR0), A[i][2] = v_{i+16} (lanes 16-31, VGPR0)
//   => D[i][j] = v_i + v_{i+16} for every column j.
// C/D layout: lane<16 holds D[r][lane] in VGPR r (rows 0-7);
//             lane>=16 holds D[r+8][lane-16] (rows 8-15).
// So lane0 sums d[0..7] = sum(v_0..7, v_16..23); lane16 sums the rest.
__global__ void finalReduceKernel(const float* __restrict__ partials,
                                  int n, float invB,
                                  float* __restrict__ out)
{
    const int lane = threadIdx.x;   // blockDim.x == 32
    float v = 0.0f;
    for (int i = lane; i < n; i += 32) v += partials[i];

    v2f a; a.x = v;    a.y = 0.0f;
    v2f b; b.x = 1.0f; b.y = 1.0f;
    v8f c = {};
    v8f d = __builtin_amdgcn_wmma_f32_16x16x4_f32(
        /*neg_a=*/false, a, /*neg_b=*/false, b,
        /*c_mod=*/(short)0, c, /*reuse_a=*/false, /*reuse_b=*/false);

    float p = d[0] + d[1] + d[2] + d[3] + d[4] + d[5] + d[6] + d[7];
    float q = __shfl_xor(p, 16, 32);   // combine lane0 (rows 0-7) with lane16 (rows 8-15)
    if (lane == 0) *out = fabsf((p + q) * invB);
}

extern "C" void kernel_launch(void* const* d_in, const int* in_sizes, int n_in,
                              void* d_out, int out_size, void* d_ws, size_t ws_size,
                              hipStream_t stream)
{
    // setup_inputs order: q_eval [B,1] (unused), reward [B], action [B, 4*m]
    const float* reward = (const float*)d_in[1];
    const float* action = (const float*)d_in[2];

    const int B  = in_sizes[1];
    const int A4 = (int)((long long)in_sizes[2] / B);
    const int m  = A4 / 4;

    float* partials = (float*)d_ws;            // nblocks floats of scratch
    const int nblocks = (B + 7) / 8;           // 8 rows (waves) per 256-thread block

    rowReduceKernel<<<nblocks, 256, 0, stream>>>(action, reward, partials, m, B);
    finalReduceKernel<<<1, 32, 0, stream>>>(partials, nblocks, 1.0f / (float)B,
                                            (float*)d_out);
}